// Helmholtz_Loss_2894807957761
// MI455X (gfx1250) — compile-verified
//
#include <hip/hip_runtime.h>
#include <cmath>

// ---------------------------------------------------------------------------
// Helmholtz loss, collapsed form:
//   per patch:  w = Q^T v (4-vector, linear in 15 corner loads)
//               patch = w^T G w  (G = M^T C_eff M, constant symmetric 4x4)
//   output   :  mean over 16*511*511 patches
// Memory-bound: 64MB input (fits in 192MB L2), ~2.7us HBM floor @ 23.3TB/s.
// 3-D grid (col-chunk, y, b): b,y block-uniform -> scalar base address, no
// integer division. Each thread covers 8 rows (y + 64k), feeding 4 chained
// V_WMMA_F32_16X16X4_F32 ops (B=ones) that accumulate 256 patch values per
// wave on the matrix pipe; the per-wave epilogue uses immediate ds_swizzle
// XOR reductions (no lane-index math).
// ---------------------------------------------------------------------------

typedef __attribute__((ext_vector_type(2))) float v2f;
typedef __attribute__((ext_vector_type(8))) float v8f;

struct HelmConsts {
    float v0, v1, v2, v3;               // column sums of M
    float g00, g11, g22, g33;           // diag of G
    float g01, g02, g03, g12, g13, g23; // off-diag of G, pre-doubled
};

#define HB 16
#define HCH 4
#define HH 512
#define HW 512
#define PH (HH - 1)              // 511 patch rows
#define PW (HW - 1)              // 511 patch cols
#define NPATCH (HB * PH * PW)    // 4,177,936
#define PLANE (HH * HW)          // 262,144

#define GX 2                     // column chunks of 256
#define GY 64                    // y in [0,64); rows covered: y + 64k, k=0..7
#define NPART (GX * GY * HB)     // 2048 block partials

__device__ __forceinline__ float patch_val(const float* __restrict__ base,
                                           int h, int c, const HelmConsts& kc) {
    const float* p0 = base + h * HW + c;   // base is block-uniform
    const float* p1 = p0 + PLANE;
    const float* p2 = p0 + 2 * PLANE;
    const float* p3 = p0 + 3 * PLANE;
    float tl0 = p0[0], tr0 = p0[1], bl0 = p0[HW], br0 = p0[HW + 1];
    float tr1 = p1[1], bl1 = p1[HW], br1 = p1[HW + 1];
    float tl2 = p2[0], tr2 = p2[1], bl2 = p2[HW], br2 = p2[HW + 1];
    float tl3 = p3[0], tr3 = p3[1], bl3 = p3[HW], br3 = p3[HW + 1];
    float w0 = kc.v0 * tl0 + kc.v1 * bl0 + kc.v2 * tl3 + kc.v3 * bl1;
    float w1 = kc.v0 * tr0 + kc.v1 * br0 + kc.v2 * tr1 + kc.v3 * br1;
    float w2 = kc.v0 * tl2 + kc.v1 * bl2 + kc.v2 * tl3 + kc.v3 * bl3;
    float w3 = kc.v0 * tr2 + kc.v1 * br2 + kc.v2 * tr3 + kc.v3 * br3;
    return kc.g00 * w0 * w0 + kc.g11 * w1 * w1 + kc.g22 * w2 * w2 + kc.g33 * w3 * w3
         + kc.g01 * (w0 * w1) + kc.g02 * (w0 * w2) + kc.g03 * (w0 * w3)
         + kc.g12 * (w1 * w2) + kc.g13 * (w1 * w3) + kc.g23 * (w2 * w3);
}

// XOR-butterfly wave32 reduction via immediate ds_swizzle (src = lane ^ mask).
__device__ __forceinline__ float wave_reduce_add(float s) {
    s += __int_as_float(__builtin_amdgcn_ds_swizzle(__float_as_int(s), 0x401f)); // ^16
    s += __int_as_float(__builtin_amdgcn_ds_swizzle(__float_as_int(s), 0x201f)); // ^8
    s += __int_as_float(__builtin_amdgcn_ds_swizzle(__float_as_int(s), 0x101f)); // ^4
    s += __int_as_float(__builtin_amdgcn_ds_swizzle(__float_as_int(s), 0x081f)); // ^2
    s += __int_as_float(__builtin_amdgcn_ds_swizzle(__float_as_int(s), 0x041f)); // ^1
    return s;
}

// Stage 1: 4 chained WMMAs per thread; EXEC is all-ones at every WMMA
// (guards wrap only the loads; invalid slots contribute 0).
__global__ void __launch_bounds__(256)
helm_partial(const float* __restrict__ x, float* __restrict__ part, HelmConsts kc) {
    const int c = blockIdx.x * 256 + threadIdx.x; // patch column
    const int y = blockIdx.y;                     // base row slot, [0,64)
    const float* base = x + (size_t)blockIdx.z * (HCH * PLANE);

    v2f ones = {1.0f, 1.0f};
    v8f acc = {};
#pragma unroll
    for (int j = 0; j < 4; ++j) {
        const int ha = y + 64 * (2 * j);          // <= 447 : always valid row
        const int hb = y + 64 * (2 * j + 1);      // only j==3,y==63 -> 511
        float pa = 0.0f, pb = 0.0f;
        if (c < PW) {
            pa = patch_val(base, ha, c, kc);
            if (hb < PH) pb = patch_val(base, hb, c, kc);
        }
        v2f a = {pa, pb};
        // D[m][n] = sum_k A[m][k] + C : matrix-pipe reduction, 64 values/op
        acc = __builtin_amdgcn_wmma_f32_16x16x4_f32(
            false, a, false, ones, (short)0, acc, false, false);
    }

    // Full sum over the 16x16 D tile counts every patch value exactly 16x
    // (once per column), independent of the C/D lane layout.
    float s = acc[0] + acc[1] + acc[2] + acc[3] + acc[4] + acc[5] + acc[6] + acc[7];
    s = wave_reduce_add(s) * 0.0625f;

    __shared__ float wpart[8];
    int wave = threadIdx.x >> 5;
    if ((threadIdx.x & 31) == 0) wpart[wave] = s;
    __syncthreads();
    if (threadIdx.x == 0) {
        float bsum = 0.0f;
        for (int wv = 0; wv < 8; ++wv) bsum += wpart[wv];
        part[blockIdx.x + GX * (blockIdx.y + GY * blockIdx.z)] = bsum;
    }
}

// Stage 2: one wave, fixed-order deterministic reduction of block partials.
__global__ void helm_final(const float* __restrict__ part, float* __restrict__ out,
                           int nPart, double invN) {
    int lane = threadIdx.x;
    float s = 0.0f;
    for (int k = lane; k < nPart; k += 32) s += part[k];
    s = wave_reduce_add(s);
    if (lane == 0) out[0] = (float)((double)s * invN);
}

// ---------------------------------------------------------------------------
// Host-side constant folding (pure math; graph-capture safe). Base matrices
// quantized to f32 to mirror the reference's float32 arrays, combined in f64.
// ---------------------------------------------------------------------------
static HelmConsts make_consts() {
    double lm[7];
    {
        double L = 0.01, p = 1.0;
        for (int i = 0; i < 7; ++i) { p *= L; lm[i] = p; }
    }
    double C1[4][4] = {
        {lm[0],     lm[1]/2.0, lm[2]/3.0, lm[3]/4.0},
        {lm[1]/2.0, lm[2]/3.0, lm[3]/4.0, lm[4]/5.0},
        {lm[2]/3.0, lm[3]/4.0, lm[4]/5.0, lm[5]/6.0},
        {lm[3]/4.0, lm[4]/5.0, lm[5]/6.0, lm[6]/7.0}};
    double C2[4][4] = {
        {0,0,0,0},{0,0,0,0},
        {0,0,4.0*lm[0],6.0*lm[1]},
        {0,0,6.0*lm[1],12.0*lm[2]}};
    double C3[4][4] = {
        {0,0,0,0},{0,0,0,0},
        {2.0*lm[0], lm[1],      (2.0/3.0)*lm[2], 0.5*lm[3]},
        {3.0*lm[1], 2.0*lm[2],  1.5*lm[3],       1.2*lm[4]}};
    double M[4][4] = {
        {1,0,0,0},{0,0,1,0},
        {-3.0/lm[1],  3.0/lm[1], -2.0/lm[0], -1.0/lm[0]},
        { 2.0/lm[2], -2.0/lm[2],  1.0/lm[1],  1.0/lm[1]}};
    for (int i = 0; i < 4; ++i)
        for (int j = 0; j < 4; ++j) {
            C1[i][j] = (double)(float)C1[i][j];
            C2[i][j] = (double)(float)C2[i][j];
            C3[i][j] = (double)(float)C3[i][j];
            M[i][j]  = (double)(float)M[i][j];
        }
    double s1 = 0, s2 = 0, s3 = 0;
    for (int i = 0; i < 4; ++i)
        for (int j = 0; j < 4; ++j) { s1 += C1[i][j]; s2 += C2[i][j]; s3 += C3[i][j]; }
    s1 = (double)(float)s1; s2 = (double)(float)s2; s3 = (double)(float)s3;

    double k2 = 20.0 * 20.0;
    double Ce[4][4];
    for (int j = 0; j < 4; ++j)
        for (int k = 0; k < 4; ++k)
            Ce[j][k] = s2 * C1[j][k] + s1 * C2[j][k] + (k2 * k2 * s1) * C1[j][k]
                     + 2.0 * s3 * C3[k][j]            // quad(C3^T)
                     + (2.0 * k2 * s1) * C3[j][k]
                     + (2.0 * k2 * s3) * C1[j][k];
    double G[4][4] = {{0}};
    for (int m = 0; m < 4; ++m)
        for (int n = 0; n < 4; ++n)
            for (int j = 0; j < 4; ++j)
                for (int k = 0; k < 4; ++k)
                    G[m][n] += M[j][m] * Ce[j][k] * M[k][n];
    double v[4] = {0, 0, 0, 0};
    for (int j = 0; j < 4; ++j)
        for (int i = 0; i < 4; ++i) v[j] += M[i][j];

    auto Gs = [&](int m, int n) { return 0.5 * (G[m][n] + G[n][m]); };
    HelmConsts kc;
    kc.v0 = (float)v[0]; kc.v1 = (float)v[1]; kc.v2 = (float)v[2]; kc.v3 = (float)v[3];
    kc.g00 = (float)Gs(0, 0); kc.g11 = (float)Gs(1, 1);
    kc.g22 = (float)Gs(2, 2); kc.g33 = (float)Gs(3, 3);
    kc.g01 = (float)(2.0 * Gs(0, 1)); kc.g02 = (float)(2.0 * Gs(0, 2));
    kc.g03 = (float)(2.0 * Gs(0, 3)); kc.g12 = (float)(2.0 * Gs(1, 2));
    kc.g13 = (float)(2.0 * Gs(1, 3)); kc.g23 = (float)(2.0 * Gs(2, 3));
    return kc;
}

extern "C" void kernel_launch(void* const* d_in, const int* in_sizes, int n_in,
                              void* d_out, int out_size, void* d_ws, size_t ws_size,
                              hipStream_t stream) {
    const float* x = (const float*)d_in[0];
    float* out = (float*)d_out;
    float* part = (float*)d_ws;  // NPART floats of scratch

    HelmConsts kc = make_consts();

    dim3 grid(GX, GY, HB);       // (col-chunk, row-slot, batch)
    helm_partial<<<grid, 256, 0, stream>>>(x, part, kc);
    helm_final<<<1, 32, 0, stream>>>(part, out, NPART, 1.0 / (double)NPATCH);
}